// SliceFineLiMEEmbedding_17325898072235
// MI455X (gfx1250) — compile-verified
//
#include <hip/hip_runtime.h>
#include <math.h>

// ---------------- problem constants ----------------
namespace {
constexpr int   kB   = 4;
constexpr int   kT   = 4096;
constexpr int   kH   = 2048;
constexpr int   kE   = 64;
constexpr int   kR   = 64;     // RANK
constexpr int   kNG  = 4;      // NGRAM
constexpr int   kNA  = kT / kNG;      // 1024 anchors per batch
constexpr int   kRows = kB * kNA;     // 4096 anchor rows
constexpr float kScaling = 16.0f / 64.0f;  // ALPHA / RANK = 0.25
constexpr float kEps = 1e-6f;

// workspace layout (float offsets)
constexpr int WS_SCALE = 0;                 // 1 float (max|h| as bits)
constexpr int WS_PBAR  = 16;                // 64 floats: sum of probs per e
constexpr int WS_LOAD  = 96;                // 64 floats: selection counts per e
constexpr int WS_W     = 256;               // 4096x64 weights
constexpr int WS_PM    = WS_W + kRows * kE; // 4096x64 p_mix_anchor
} // namespace

typedef __attribute__((ext_vector_type(2))) float v2f;
typedef __attribute__((ext_vector_type(4))) float v4f;
typedef __attribute__((ext_vector_type(8))) float v8f;

// ---------------- kernel 0: zero accumulators ----------------
__global__ __launch_bounds__(256) void init_kernel(float* __restrict__ wsf) {
  wsf[threadIdx.x] = 0.0f;   // covers WS_SCALE, WS_PBAR, WS_LOAD
}

// ---------------- kernel 1: global max |h_anchor| ----------------
__global__ __launch_bounds__(64) void scale_kernel(const int* __restrict__ ids,
                                                   const float* __restrict__ emb,
                                                   float* __restrict__ wsf) {
  __shared__ float red[64];
  const int row = blockIdx.x;           // b*kNA + a
  const int b   = row >> 10;
  const int a   = row & (kNA - 1);
  const int e   = threadIdx.x;
  const int token = ids[b * kT + a * kNG + (kNG - 1)];
  float v = fabsf(emb[(long long)token * kH + e]);
  red[e] = v; __syncthreads();
  for (int s = 32; s; s >>= 1) { if (e < s) red[e] = fmaxf(red[e], red[e + s]); __syncthreads(); }
  if (e == 0) atomicMax((unsigned int*)&wsf[WS_SCALE], __float_as_uint(red[0]));
}

// ---------------- kernel 2: softmax + auto-topk ----------------
__global__ __launch_bounds__(64) void softmax_topk_kernel(const int* __restrict__ ids,
                                                          const float* __restrict__ emb,
                                                          float* __restrict__ wsf) {
  __shared__ float red[64];
  __shared__ float bc;
  const int row = blockIdx.x;
  const int b   = row >> 10;
  const int a   = row & (kNA - 1);
  const int e   = threadIdx.x;
  const int token = ids[b * kT + a * kNG + (kNG - 1)];

  const float scale = fmaxf(__uint_as_float(((const unsigned int*)wsf)[WS_SCALE]), kEps);
  const float x = emb[(long long)token * kH + e] * (1.0f / scale);  // TEMP == 1

  // max logit
  red[e] = x; __syncthreads();
  for (int s = 32; s; s >>= 1) { if (e < s) red[e] = fmaxf(red[e], red[e + s]); __syncthreads(); }
  if (e == 0) bc = red[0];
  __syncthreads();
  const float xmax = bc;
  __syncthreads();

  // sum of exp
  const float ex = __expf(x - xmax);
  red[e] = ex; __syncthreads();
  for (int s = 32; s; s >>= 1) { if (e < s) red[e] += red[e + s]; __syncthreads(); }
  if (e == 0) bc = red[0];
  __syncthreads();
  const float sum = bc;
  __syncthreads();

  const float p    = ex / sum;
  const float pmax = 1.0f / sum;          // == max(probs), exact (ex==1 at argmax)
  const float thr  = 0.5f * pmax;         // THRESH * max_prob
  const float mask = 1.0f / (1.0f + __expf(-(p - thr) * 2.0f));  // 1/STEMP = 2
  const float wu   = p * mask;

  red[e] = wu; __syncthreads();
  for (int s = 32; s; s >>= 1) { if (e < s) red[e] += red[e + s]; __syncthreads(); }
  if (e == 0) bc = red[0];
  __syncthreads();
  const float w = wu / (bc + 1e-9f);

  wsf[WS_W + (long long)row * kE + e] = w;
  atomicAdd(&wsf[WS_PBAR + e], p);
  atomicAdd(&wsf[WS_LOAD + e], (p >= thr) ? 1.0f : 0.0f);
}

// ---------------- kernel 3: WMMA f32 GEMM: PM = W(4096x64) x L(64x64) ----
// One wave per 16x16 output tile; K-loop over E=64 in steps of 4 using
// V_WMMA_F32_16X16X4_F32. A: lane(l&15)=M, (l>>4)*2 = K-pair; B mirrored.
__global__ __launch_bounds__(256) void gemm_wmma_kernel(const float* __restrict__ W,
                                                        const float* __restrict__ L,
                                                        float* __restrict__ PM) {
  const int wave = threadIdx.x >> 5;
  const int lane = threadIdx.x & 31;
  const int tile = blockIdx.x * 8 + wave;   // 1024 tiles total
  const int tileRow = tile >> 2;            // 256 row-tiles of 16 anchor rows
  const int tileCol = tile & 3;             // 4 col-tiles of 16 rank cols
  const int m  = lane & 15;
  const int kp = (lane >> 4) << 1;          // 0 or 2

  const float* wrow = W + (long long)(tileRow * 16 + m) * kE;
  const float* lcol = L + tileCol * 16 + (lane & 15);

  v8f c = {};
#pragma unroll
  for (int k = 0; k < kE; k += 4) {
    v2f av, bv;
    av.x = wrow[k + kp];
    av.y = wrow[k + kp + 1];
    bv.x = lcol[(k + kp) * kR];
    bv.y = lcol[(k + kp + 1) * kR];
    c = __builtin_amdgcn_wmma_f32_16x16x4_f32(
        /*neg_a=*/false, av, /*neg_b=*/false, bv,
        /*c_mod=*/(short)0, c, /*reuse_a=*/false, /*reuse_b=*/false);
  }

  // D layout: VGPR v -> M = v + 8*(lane>=16), N = lane&15
  const int mbase = (lane >> 4) * 8;
  const int n = lane & 15;
  float* outp = PM + tileCol * 16 + n;
#pragma unroll
  for (int v = 0; v < 8; ++v)
    outp[(long long)(tileRow * 16 + mbase + v) * kR] = c[v];
}

// ---------------- kernel 4: fused gather + low-rank add + store ----------
__global__ __launch_bounds__(256) void out_kernel(const int* __restrict__ ids,
                                                  const float* __restrict__ emb,
                                                  const float* __restrict__ part,
                                                  const float* __restrict__ PM,
                                                  float* __restrict__ out) {
  const long long row = blockIdx.x;       // b*kT + t
  const int t = (int)(row & (kT - 1));
  const int b = (int)(row >> 12);         // kT = 2^12
  const int token = ids[row];

  const v4f* __restrict__ src = (const v4f*)(emb + (long long)token * kH);
  v4f* __restrict__ dst = (v4f*)(out + row * (long long)kH);
  const float* __restrict__ pm = PM + ((long long)b * kNA + (t >> 2)) * kR;
  const float* __restrict__ dl = part + (long long)token * kR;

  const int tid = threadIdx.x;
#pragma unroll
  for (int j = 0; j < 2; ++j) {
    const int v4 = tid + j * 256;         // 512 float4 per row
    v4f val = src[v4];
    const int f0 = v4 * 4;
    if (f0 < kR) {
      val.x += dl[f0 + 0] * pm[f0 + 0] * kScaling;
      val.y += dl[f0 + 1] * pm[f0 + 1] * kScaling;
      val.z += dl[f0 + 2] * pm[f0 + 2] * kScaling;
      val.w += dl[f0 + 3] * pm[f0 + 3] * kScaling;
    }
    __builtin_nontemporal_store(val, &dst[v4]);  // streaming 128MB write
  }
}

// ---------------- kernel 5: scalar stats ----------------
__global__ __launch_bounds__(64) void stats_kernel(const float* __restrict__ wsf,
                                                   float* __restrict__ out3) {
  __shared__ float red[64];
  const int e = threadIdx.x;
  const float pbar = wsf[WS_PBAR + e] * (1.0f / (float)kRows);
  const float ld   = wsf[WS_LOAD + e];

  float s_pb2 = 0.f, s_ld = 0.f, s_ld2 = 0.f, s_kl = 0.f;

  red[e] = pbar * pbar; __syncthreads();
  for (int s = 32; s; s >>= 1) { if (e < s) red[e] += red[e + s]; __syncthreads(); }
  if (e == 0) s_pb2 = red[0];
  __syncthreads();

  red[e] = ld; __syncthreads();
  for (int s = 32; s; s >>= 1) { if (e < s) red[e] += red[e + s]; __syncthreads(); }
  if (e == 0) s_ld = red[0];
  __syncthreads();

  red[e] = ld * ld; __syncthreads();
  for (int s = 32; s; s >>= 1) { if (e < s) red[e] += red[e + s]; __syncthreads(); }
  if (e == 0) s_ld2 = red[0];
  __syncthreads();

  red[e] = pbar * (logf(fmaxf(pbar, 1e-8f)) + logf(64.0f)); __syncthreads();
  for (int s = 32; s; s >>= 1) { if (e < s) red[e] += red[e + s]; __syncthreads(); }
  if (e == 0) s_kl = red[0];

  if (e == 0) {
    const float imp = 64.0f * s_pb2 - 1.0f;
    const float den = fmaxf(s_ld, 1.0f);
    const float scv = 64.0f * s_ld2 / (den * den) - 1.0f;
    out3[0] = imp;
    out3[1] = scv;
    out3[2] = s_kl;
  }
}

// ---------------- launch ----------------
extern "C" void kernel_launch(void* const* d_in, const int* in_sizes, int n_in,
                              void* d_out, int out_size, void* d_ws, size_t ws_size,
                              hipStream_t stream) {
  (void)in_sizes; (void)n_in; (void)out_size; (void)ws_size;
  const int*   ids   = (const int*)d_in[0];
  const float* emb   = (const float*)d_in[1];
  const float* part  = (const float*)d_in[2];
  const float* limes = (const float*)d_in[3];
  float* out = (float*)d_out;
  float* wsf = (float*)d_ws;

  init_kernel<<<1, 256, 0, stream>>>(wsf);
  scale_kernel<<<kRows, 64, 0, stream>>>(ids, emb, wsf);
  softmax_topk_kernel<<<kRows, 64, 0, stream>>>(ids, emb, wsf);
  gemm_wmma_kernel<<<(kRows / 16) * (kR / 16) / 8, 256, 0, stream>>>(
      wsf + WS_W, limes, wsf + WS_PM);
  out_kernel<<<kB * kT, 256, 0, stream>>>(ids, emb, part, wsf + WS_PM, out);
  stats_kernel<<<1, 64, 0, stream>>>(wsf, out + (long long)kB * kT * kH);
}